// Generator_83786222010386
// MI455X (gfx1250) — compile-verified
//
#include <hip/hip_runtime.h>
#include <hip/hip_bf16.h>
#include <math.h>

typedef __attribute__((ext_vector_type(2))) float v2f;
typedef __attribute__((ext_vector_type(8))) float v8f;

#define TOKS 12800
#define SEQL 200
#define NB   64

__device__ __forceinline__ v8f wmma4(v2f a, v2f b, v8f c) {
    // D = A(16x4 f32) * B(4x16 f32) + C(16x16 f32)
    return __builtin_amdgcn_wmma_f32_16x16x4_f32(false, a, false, b, (short)0, c, false, false);
}

__device__ __forceinline__ v8f vzero8() {
    v8f c; for (int v = 0; v < 8; ++v) c[v] = 0.0f; return c;
}

// branch-free transcendental-unit forms: v_exp_f32 + v_rcp_f32 (both TRANS ops
// that co-execute with WMMA); avoids the precise-division v_div_scale sequence.
__device__ __forceinline__ float fast_rcp(float x) {
    return __builtin_amdgcn_rcpf(x);
}
__device__ __forceinline__ float fast_sigmoid(float x) {
    return fast_rcp(1.0f + __expf(-x));
}
__device__ __forceinline__ float fast_tanh(float x) {
    return 1.0f - 2.0f * fast_rcp(1.0f + __expf(2.0f * x));   // exact at +/-inf
}

// ---------------------------------------------------------------------------
// K1: embedding gather -> tanh(z@W0+b0)@W1+b1 -> xp = mlp_out@Wih + bih
// one block = 16 tokens x one stream. 256 threads = 8 waves.
// ---------------------------------------------------------------------------
__global__ __launch_bounds__(256) void k_embed_mlp_xp(
    const int* __restrict__ sidx, const int* __restrict__ ridx,
    const float* __restrict__ emb,
    const float* __restrict__ w0, const float* __restrict__ b0,
    const float* __restrict__ w1, const float* __restrict__ b1m,
    const float* __restrict__ wih, const float* __restrict__ bih,
    float* __restrict__ xpS, float* __restrict__ xpR)
{
    __shared__ float z[16][130];
    __shared__ float h1[16][130];
    __shared__ float h2[16][130];
    const int m0   = blockIdx.x * 16;
    const int strm = blockIdx.y;
    const int* idx = strm ? ridx : sidx;
    float* xp      = strm ? xpR  : xpS;
    const int tid = threadIdx.x;

    for (int i = tid; i < 16 * 128; i += 256) {
        int row = i >> 7, col = i & 127;
        z[row][col] = emb[(long)idx[m0 + row] * 128 + col];
    }
    __syncthreads();

    const int wave = tid >> 5, lane = tid & 31;
    const int r = lane & 15, half = lane >> 4;

    { // layer 1: h1 = tanh(z @ w0 + b0), wave -> n-tile
        int n0 = wave * 16;
        v8f c = vzero8();
#pragma unroll 4
        for (int k = 0; k < 128; k += 4) {
            v2f a, bv;
            a.x  = z[r][k + 2 * half];
            a.y  = z[r][k + 2 * half + 1];
            bv.x = w0[(k + 2 * half) * 128 + n0 + r];
            bv.y = w0[(k + 2 * half + 1) * 128 + n0 + r];
            c = wmma4(a, bv, c);
        }
        float bb = b0[n0 + r];
        for (int v = 0; v < 8; ++v) h1[v + 8 * half][n0 + r] = fast_tanh(c[v] + bb);
    }
    __syncthreads();

    { // layer 2: h2 = h1 @ w1 + b1
        int n0 = wave * 16;
        v8f c = vzero8();
#pragma unroll 4
        for (int k = 0; k < 128; k += 4) {
            v2f a, bv;
            a.x  = h1[r][k + 2 * half];
            a.y  = h1[r][k + 2 * half + 1];
            bv.x = w1[(k + 2 * half) * 128 + n0 + r];
            bv.y = w1[(k + 2 * half + 1) * 128 + n0 + r];
            c = wmma4(a, bv, c);
        }
        float bb = b1m[n0 + r];
        for (int v = 0; v < 8; ++v) h2[v + 8 * half][n0 + r] = c[v] + bb;
    }
    __syncthreads();

    // xp = h2 @ wih + bih  (N = 384 -> 24 n-tiles, 3 per wave)
    for (int nt = wave; nt < 24; nt += 8) {
        int n0 = nt * 16;
        v8f c = vzero8();
#pragma unroll 4
        for (int k = 0; k < 128; k += 4) {
            v2f a, bv;
            a.x  = h2[r][k + 2 * half];
            a.y  = h2[r][k + 2 * half + 1];
            bv.x = wih[(k + 2 * half) * 384 + n0 + r];
            bv.y = wih[(k + 2 * half + 1) * 384 + n0 + r];
            c = wmma4(a, bv, c);
        }
        float bb = bih[n0 + r];
        for (int v = 0; v < 8; ++v)
            xp[(long)(m0 + v + 8 * half) * 384 + n0 + r] = c[v] + bb;
    }
}

// ---------------------------------------------------------------------------
// K2: GRU recurrence, batch-parallel: block = 16 batches x one stream
// (grid 4x2 = 8 WGPs). 256 threads = 8 waves, 3 n-tiles per wave.
// whh B-tiles are invariant across all 200 steps -> preloaded into registers
// (3 tiles x 32 k-steps x v2f = 192 VGPRs/lane); recurrent inner loop is pure
// LDS-A + register-B WMMA with no global loads on the critical path.
// ---------------------------------------------------------------------------
__global__ __launch_bounds__(256) void k_gru(
    const float* __restrict__ xpS, const float* __restrict__ xpR,
    const float* __restrict__ whh, const float* __restrict__ bhh,
    float* __restrict__ hsS, float* __restrict__ hsR)
{
    __shared__ float h[16][130];
    __shared__ float gh[16][384];
    const int bg0  = blockIdx.x * 16;      // batch group
    const int strm = blockIdx.y;
    const float* xp = strm ? xpR : xpS;
    float* hs       = strm ? hsR : hsS;
    const int tid = threadIdx.x;
    const int wave = tid >> 5, lane = tid & 31;
    const int r = lane & 15, half = lane >> 4;

    // preload whh tiles for this wave's 3 n-tiles
    v2f barr[96];
#pragma unroll
    for (int i = 0; i < 3; ++i) {
        int n0 = (wave * 3 + i) * 16;
#pragma unroll
        for (int ks = 0; ks < 32; ++ks) {
            int k = ks * 4;
            v2f bv;
            bv.x = whh[(k + 2 * half) * 384 + n0 + r];
            bv.y = whh[(k + 2 * half + 1) * 384 + n0 + r];
            barr[i * 32 + ks] = bv;
        }
    }

    for (int i = tid; i < 16 * 130; i += 256) (&h[0][0])[i] = 0.0f;
    __syncthreads();

    for (int t = 0; t < SEQL; ++t) {
        // gh = h @ whh   (M=16, N=384, K=128)
#pragma unroll
        for (int i = 0; i < 3; ++i) {
            int n0 = (wave * 3 + i) * 16;
            v8f c = vzero8();
#pragma unroll
            for (int ks = 0; ks < 32; ++ks) {
                int k = ks * 4;
                v2f a;
                a.x = h[r][k + 2 * half];
                a.y = h[r][k + 2 * half + 1];
                c = wmma4(a, barr[i * 32 + ks], c);
            }
            for (int v = 0; v < 8; ++v)
                gh[v + 8 * half][n0 + r] = c[v];
        }
        __syncthreads();

        // gate update (torch GRU, gate order r,z,n): 16*128 elems, 8 per thread
        for (int i = tid; i < 16 * 128; i += 256) {
            int bb = i >> 7, d = i & 127;
            long xo = (long)((bg0 + bb) * SEQL + t) * 384;
            float xr = xp[xo + d], xz = xp[xo + 128 + d], xn = xp[xo + 256 + d];
            float gr = gh[bb][d]       + bhh[d];
            float gz = gh[bb][128 + d] + bhh[128 + d];
            float gn = gh[bb][256 + d] + bhh[256 + d];
            float rt = fast_sigmoid(xr + gr);
            float zt = fast_sigmoid(xz + gz);
            float nt = fast_tanh(xn + rt * gn);
            float hv = (1.0f - zt) * nt + zt * h[bb][d];
            h[bb][d] = hv;
            hs[(long)((bg0 + bb) * SEQL + t) * 128 + d] = hv;
        }
        __syncthreads();
    }
}

// ---------------------------------------------------------------------------
// K3: generic WMMA GEMM, M multiple of 16 (grid.x), K == 128, A LDS-staged.
// 128 threads = 4 waves; n-tile = blockIdx.y*4 + wave (guarded vs N).
// ---------------------------------------------------------------------------
__global__ __launch_bounds__(128) void k_gemm(
    const float* __restrict__ A, int lda,
    const float* __restrict__ B, int ldb,
    const float* __restrict__ bias,
    float* __restrict__ C, int ldc,
    int N, int K, int accum)
{
    __shared__ float As[16][130];
    const int m0 = blockIdx.x * 16;
    const int tid = threadIdx.x;
    for (int i = tid; i < 16 * 128; i += 128) {
        int row = i >> 7, col = i & 127;
        As[row][col] = A[(long)(m0 + row) * lda + col];
    }
    __syncthreads();

    const int wave = tid >> 5, lane = tid & 31;
    const int r = lane & 15, half = lane >> 4;
    const int n0 = (blockIdx.y * 4 + wave) * 16;
    if (n0 >= N) return; // wave-uniform: EXEC stays all-ones in live waves

    v8f c = vzero8();
    if (accum)
        for (int v = 0; v < 8; ++v) c[v] = C[(long)(m0 + v + 8 * half) * ldc + n0 + r];
#pragma unroll 4
    for (int k = 0; k < K; k += 4) {
        v2f a, bv;
        a.x  = As[r][k + 2 * half];
        a.y  = As[r][k + 2 * half + 1];
        bv.x = B[(k + 2 * half) * ldb + n0 + r];
        bv.y = B[(k + 2 * half + 1) * ldb + n0 + r];
        c = wmma4(a, bv, c);
    }
    float bb = bias ? bias[n0 + r] : 0.0f;
    for (int v = 0; v < 8; ++v)
        C[(long)(m0 + v + 8 * half) * ldc + n0 + r] = c[v] + bb;
}

// ---------------------------------------------------------------------------
// K4: attention core. block = (stream, batch, head, 16-query tile), 128 thr.
// scores via WMMA -> LDS -> mask/scale/reweight (reference order) -> softmax
// -> A@V via WMMA. S=200 padded to 208.
// ---------------------------------------------------------------------------
__global__ __launch_bounds__(128) void k_attn(
    const float* __restrict__ qS, const float* __restrict__ kS, const float* __restrict__ vS,
    const float* __restrict__ qR, const float* __restrict__ kR, const float* __restrict__ vR,
    const float* __restrict__ rw, float* __restrict__ attnS, float* __restrict__ attnR)
{
    __shared__ float sc[16][212];
    const int qt = blockIdx.x % 13, hd = blockIdx.x / 13;
    const int b = blockIdx.y, strm = blockIdx.z;
    const float* q = strm ? qR : qS;
    const float* kk_ = strm ? kR : kS;
    const float* vv_ = strm ? vR : vS;
    float* ao = strm ? attnR : attnS;
    const int causal = strm; // r-stream is causal

    const int tid = threadIdx.x, wave = tid >> 5, lane = tid & 31;
    const int r = lane & 15, half = lane >> 4;
    const int m0 = qt * 16;
    const long base = (long)b * SEQL * 128;
    const int hc = hd * 64;

    // scores: 13 key tiles, round-robin over 4 waves
    for (int nt = wave; nt < 13; nt += 4) {
        int n0 = nt * 16;
        v8f c = vzero8();
#pragma unroll 4
        for (int k4 = 0; k4 < 64; k4 += 4) {
            v2f a, bv;
            int ar = m0 + r;
            int kp = n0 + r;
            bool am = ar < SEQL, bm = kp < SEQL;
            a.x  = am ? q[base + (long)ar * 128 + hc + k4 + 2 * half] : 0.0f;
            a.y  = am ? q[base + (long)ar * 128 + hc + k4 + 2 * half + 1] : 0.0f;
            bv.x = bm ? kk_[base + (long)kp * 128 + hc + k4 + 2 * half] : 0.0f;
            bv.y = bm ? kk_[base + (long)kp * 128 + hc + k4 + 2 * half + 1] : 0.0f;
            c = wmma4(a, bv, c);
        }
        for (int v = 0; v < 8; ++v) sc[v + 8 * half][n0 + r] = c[v];
    }
    __syncthreads();

    // per-row softmax (16 rows handled by threads 0..15)
    if (tid < 16) {
        int qrow = m0 + tid;
        if (qrow < SEQL) {
            float mx = -1e30f;
            for (int j = 0; j < SEQL; ++j) {
                float s = sc[tid][j];
                if (causal && j > qrow) s -= 1e9f;  // mask BEFORE scale, as in source
                s *= 0.125f;                        // 1/sqrt(64)
                s *= rw[b * SEQL + j];              // per-key reweight
                sc[tid][j] = s;
                mx = fmaxf(mx, s);
            }
            float sum = 0.0f;
            for (int j = 0; j < SEQL; ++j) {
                float e = __expf(sc[tid][j] - mx);
                sc[tid][j] = e; sum += e;
            }
            float inv = fast_rcp(sum);
            for (int j = 0; j < SEQL; ++j) sc[tid][j] *= inv;
            for (int j = SEQL; j < 208; ++j) sc[tid][j] = 0.0f;
        } else {
            for (int j = 0; j < 208; ++j) sc[tid][j] = 0.0f;
        }
    }
    __syncthreads();

    // out = probs @ V : dk=64 -> 4 n-tiles, one per wave; K padded to 208
    {
        int n0 = wave * 16;
        v8f c = vzero8();
#pragma unroll 4
        for (int kq = 0; kq < 208; kq += 4) {
            v2f a, bv;
            a.x = sc[r][kq + 2 * half];
            a.y = sc[r][kq + 2 * half + 1];
            int kp = kq + 2 * half;
            bv.x = (kp     < SEQL) ? vv_[base + (long)kp * 128 + hc + n0 + r] : 0.0f;
            bv.y = (kp + 1 < SEQL) ? vv_[base + (long)(kp + 1) * 128 + hc + n0 + r] : 0.0f;
            c = wmma4(a, bv, c);
        }
        for (int v = 0; v < 8; ++v) {
            int row = m0 + v + 8 * half;
            if (row < SEQL) ao[base + (long)row * 128 + hc + n0 + r] = c[v];
        }
    }
}

// ---------------------------------------------------------------------------
// K5: final fused low-rank apply:
// out[n,o] = Sum_{k,d} t[n,k] x[n,d] C[(k*128+d),o]  +  x @ B1  +  bias2[n,o]
// block = 16 tokens, 256 threads = 8 waves (one n-tile each).
// lora_C (2 MB) is L2-resident; next 64KB panel prefetched per outer step.
// ---------------------------------------------------------------------------
__global__ __launch_bounds__(256) void k_final(
    const float* __restrict__ x, const float* __restrict__ tmat,
    const float* __restrict__ loraC, const float* __restrict__ b1v,
    const float* __restrict__ bias2, float* __restrict__ out)
{
    __shared__ float xs[16][130];
    __shared__ float ts[16][34];
    const int m0 = blockIdx.x * 16;
    const int tid = threadIdx.x;
    for (int i = tid; i < 16 * 128; i += 256) {
        int row = i >> 7, col = i & 127;
        xs[row][col] = x[(long)(m0 + row) * 128 + col];
    }
    for (int i = tid; i < 16 * 32; i += 256) {
        int row = i >> 5, col = i & 31;
        ts[row][col] = tmat[(m0 + row) * 32 + col];
    }
    __syncthreads();

    const int wave = tid >> 5, lane = tid & 31;
    const int r = lane & 15, half = lane >> 4;
    const int n0 = wave * 16;

    v8f c;
    for (int v = 0; v < 8; ++v)
        c[v] = bias2[(long)(m0 + v + 8 * half) * 128 + n0 + r];

    // main K=4096 sweep: U[n, kb*128+d] = t[n,kb] * x[n,d]
#pragma unroll 1
    for (int kq0 = 0; kq0 < 4096; kq0 += 128) {
        if (kq0 + 128 < 4096)
            __builtin_prefetch(loraC + (long)(kq0 + 128) * 128 + n0 + lane * 4, 0, 1);
        float tv = ts[r][kq0 >> 7];                 // t scalar, constant over 128-K panel
        const float* Cp = loraC + (long)kq0 * 128;
#pragma unroll
        for (int kd = 0; kd < 128; kd += 4) {
            v2f a, bv;
            a.x  = tv * xs[r][kd + 2 * half];
            a.y  = tv * xs[r][kd + 2 * half + 1];
            bv.x = Cp[(kd + 2 * half) * 128 + n0 + r];
            bv.y = Cp[(kd + 2 * half + 1) * 128 + n0 + r];
            c = wmma4(a, bv, c);
        }
    }
    // K=128 tail: x @ B1 (b1 viewed as [128][128])
#pragma unroll 4
    for (int kd = 0; kd < 128; kd += 4) {
        v2f a, bv;
        a.x  = xs[r][kd + 2 * half];
        a.y  = xs[r][kd + 2 * half + 1];
        bv.x = b1v[(kd + 2 * half) * 128 + n0 + r];
        bv.y = b1v[(kd + 2 * half + 1) * 128 + n0 + r];
        c = wmma4(a, bv, c);
    }
    for (int v = 0; v < 8; ++v)
        out[(long)(m0 + v + 8 * half) * 128 + n0 + r] = c[v];
}

// ---------------------------------------------------------------------------
extern "C" void kernel_launch(void* const* d_in, const int* in_sizes, int n_in,
                              void* d_out, int out_size, void* d_ws, size_t ws_size,
                              hipStream_t stream)
{
    (void)in_sizes; (void)n_in; (void)out_size; (void)ws_size;
    const float* x       = (const float*)d_in[0];
    const int*   ridx    = (const int*)  d_in[1];
    const int*   sidx    = (const int*)  d_in[2];
    const float* rw      = (const float*)d_in[3];
    const float* emb     = (const float*)d_in[4];
    const float* mlp_w0  = (const float*)d_in[5];
    const float* mlp_b0  = (const float*)d_in[6];
    const float* mlp_w1  = (const float*)d_in[7];
    const float* mlp_b1  = (const float*)d_in[8];
    const float* gru_wih = (const float*)d_in[9];
    const float* gru_whh = (const float*)d_in[10];
    const float* gru_bih = (const float*)d_in[11];
    const float* gru_bhh = (const float*)d_in[12];
    const float* saaq_w  = (const float*)d_in[13];
    const float* saaq_b  = (const float*)d_in[14];
    const float* saar_w  = (const float*)d_in[15];
    const float* saar_b  = (const float*)d_in[16];
    const float* lora_B  = (const float*)d_in[17];
    const float* lora_C  = (const float*)d_in[18];
    const float* b1      = (const float*)d_in[19];
    const float* w2      = (const float*)d_in[20];
    const float* b2      = (const float*)d_in[21];
    float* out = (float*)d_out;

    // workspace layout (floats), with phase-based overlays
    float* ws  = (float*)d_ws;
    float* xpS = ws;                        // 12800*384
    float* xpR = xpS + (long)TOKS * 384;    // 12800*384
    float* hsS = xpR + (long)TOKS * 384;    // 12800*128
    float* hsR = hsS + (long)TOKS * 128;    // 12800*128
    // after GRU, xp region is dead -> reuse for q/k/v (6 * 12800*128 == 2 * 12800*384)
    float* qS = xpS;
    float* kS = qS + (long)TOKS * 128;
    float* vS = kS + (long)TOKS * 128;
    float* qR = vS + (long)TOKS * 128;
    float* kR = qR + (long)TOKS * 128;
    float* vR = kR + (long)TOKS * 128;
    // after QKV GEMMs, hs region is dead -> attention outputs
    float* attnS = hsS;
    float* attnR = hsR;
    // after attention core, qkv region is dead -> h-sum, t, bias2
    float* hsum  = xpS;
    float* tmat  = hsum + (long)TOKS * 128;
    float* bias2 = tmat + (long)TOKS * 32;

    // 1) embed -> MLP -> GRU input projection (both streams)
    k_embed_mlp_xp<<<dim3(TOKS / 16, 2), 256, 0, stream>>>(
        sidx, ridx, emb, mlp_w0, mlp_b0, mlp_w1, mlp_b1, gru_wih, gru_bih, xpS, xpR);

    // 2) GRU recurrence: batch-parallel, 4 batch-groups x 2 streams
    k_gru<<<dim3(4, 2), 256, 0, stream>>>(xpS, xpR, gru_whh, gru_bhh, hsS, hsR);

    // 3) QKV projections (stream s: saaq; stream r: saar)
    k_gemm<<<dim3(TOKS / 16, 2), 128, 0, stream>>>(hsS, 128, saaq_w + 0 * 16384, 128, saaq_b + 0 * 128, qS, 128, 128, 128, 0);
    k_gemm<<<dim3(TOKS / 16, 2), 128, 0, stream>>>(hsS, 128, saaq_w + 1 * 16384, 128, saaq_b + 1 * 128, kS, 128, 128, 128, 0);
    k_gemm<<<dim3(TOKS / 16, 2), 128, 0, stream>>>(hsS, 128, saaq_w + 2 * 16384, 128, saaq_b + 2 * 128, vS, 128, 128, 128, 0);
    k_gemm<<<dim3(TOKS / 16, 2), 128, 0, stream>>>(hsR, 128, saar_w + 0 * 16384, 128, saar_b + 0 * 128, qR, 128, 128, 128, 0);
    k_gemm<<<dim3(TOKS / 16, 2), 128, 0, stream>>>(hsR, 128, saar_w + 1 * 16384, 128, saar_b + 1 * 128, kR, 128, 128, 128, 0);
    k_gemm<<<dim3(TOKS / 16, 2), 128, 0, stream>>>(hsR, 128, saar_w + 2 * 16384, 128, saar_b + 2 * 128, vR, 128, 128, 128, 0);

    // 4) attention cores (13 q-tiles x 2 heads, 64 batches, 2 streams)
    k_attn<<<dim3(26, NB, 2), 128, 0, stream>>>(qS, kS, vS, qR, kR, vR, rw, attnS, attnR);

    // 5) output projections, accumulated: h = attnS@Wo_s+bo_s + attnR@Wo_r+bo_r
    k_gemm<<<dim3(TOKS / 16, 2), 128, 0, stream>>>(attnS, 128, saaq_w + 3 * 16384, 128, saaq_b + 3 * 128, hsum, 128, 128, 128, 0);
    k_gemm<<<dim3(TOKS / 16, 2), 128, 0, stream>>>(attnR, 128, saar_w + 3 * 16384, 128, saar_b + 3 * 128, hsum, 128, 128, 128, 1);

    // 6) t = h @ lora_B  (N=32), bias2 = h @ w2 + b2
    k_gemm<<<dim3(TOKS / 16, 1), 128, 0, stream>>>(hsum, 128, lora_B, 32, nullptr, tmat, 32, 32, 128, 0);
    k_gemm<<<dim3(TOKS / 16, 2), 128, 0, stream>>>(hsum, 128, w2, 128, b2, bias2, 128, 128, 128, 0);

    // 7) fused low-rank weight apply
    k_final<<<TOKS / 16, 256, 0, stream>>>(x, tmat, lora_C, b1, bias2, out);
}